// PhaseAdaptiveInput_46557445489275
// MI455X (gfx1250) — compile-verified
//
#include <hip/hip_runtime.h>
#include <stdint.h>
#include <stddef.h>

#ifndef __has_builtin
#define __has_builtin(x) 0
#endif

// ---- problem constants (fixed by the reference) ----
#define APF 22            // active features per sample (N_GROUPS)
#define LPA 64            // output slice width
#define OUT_DIM 640       // weight row stride (LPA * COUNT)
#define BUCKET_SIZE 6     // MAX_PLY / COUNT
#define ROW_BYTES 256     // LPA * 4 bytes staged per gathered row
#define SAMPLE_BYTES (APF * ROW_BYTES)   // 5632 B staged per sample
#define WAVES_PER_BLOCK 4
#define THREADS (WAVES_PER_BLOCK * 32)
#define BUF_BYTES (2 * SAMPLE_BYTES)     // double buffer per wave
#define ASYNC_PER_SAMPLE 11              // 22 rows * 16 b128 chunks / 32 lanes
#define SAMPLES_PER_WAVE 2               // keeps pipeline + maximizes wave count

// Pointer types matching the builtin's signature: v4i in AS1 (global) / AS3 (LDS).
typedef int v4i __attribute__((ext_vector_type(4)));
typedef __attribute__((address_space(1))) v4i* gptr_v4i;
typedef __attribute__((address_space(3))) v4i* lptr_v4i;

#if __has_builtin(__builtin_amdgcn_global_load_async_to_lds_b128)
#define HAVE_ASYNC_BUILTIN 1
#endif
#if __has_builtin(__builtin_amdgcn_s_wait_asynccnt)
#define HAVE_WAITASYNC_BUILTIN 1
#endif

// One 16-byte async DMA gather: global -> LDS, tracked on ASYNCcnt.
__device__ __forceinline__ void async_copy_b128(const float* g, char* lds_generic) {
  uint32_t lds_off = (uint32_t)(uintptr_t)lds_generic;  // low 32 bits of generic LDS addr == DS offset
#ifdef HAVE_ASYNC_BUILTIN
  __builtin_amdgcn_global_load_async_to_lds_b128(
      (gptr_v4i)(uintptr_t)g, (lptr_v4i)(uintptr_t)lds_off, /*offset=*/0, /*cpol=*/0);
#else
  unsigned long long ga = (unsigned long long)(uintptr_t)g;
  asm volatile("global_load_async_to_lds_b128 %0, %1, off"
               :: "v"(lds_off), "v"(ga) : "memory");
#endif
}

__device__ __forceinline__ void wait_async_all() {
#ifdef HAVE_WAITASYNC_BUILTIN
  __builtin_amdgcn_s_wait_asynccnt(0);
#else
  asm volatile("s_wait_asynccnt 0" ::: "memory");
#endif
  asm volatile("" ::: "memory");  // compiler fence: don't hoist LDS reads above the wait
}

__device__ __forceinline__ void wait_async_keep_one_sample() {
  // 11 newest ops (next sample's prefetch) may remain in flight; async loads
  // complete in order, so ASYNCcnt <= 11 implies the current buffer is ready.
#ifdef HAVE_WAITASYNC_BUILTIN
  __builtin_amdgcn_s_wait_asynccnt(ASYNC_PER_SAMPLE);
#else
  asm volatile("s_wait_asynccnt 11" ::: "memory");
#endif
  asm volatile("" ::: "memory");
}

__global__ __launch_bounds__(THREADS) void PhaseAdaptiveInput_kernel(
    const int* __restrict__ cols,      // [m*APF] feature column ids (row 1 of feature_indices)
    const float* __restrict__ values,  // [m*APF]
    const int* __restrict__ ply,       // [m]
    const float* __restrict__ weight,  // [N_FEATS, OUT_DIM]
    const float* __restrict__ bias,    // [OUT_DIM]
    float* __restrict__ out,           // [m, LPA]
    int m, int total_waves)
{
  __shared__ alignas(16) char smem[WAVES_PER_BLOCK * BUF_BYTES];
  const int lane = threadIdx.x & 31;
  const int wid  = threadIdx.x >> 5;
  const int gw   = blockIdx.x * WAVES_PER_BLOCK + wid;
  char* wbuf = smem + wid * BUF_BYTES;   // private double buffer for this wave

  if (gw >= m) return;

  const float QS = 127.0f;
  const float QI = 1.0f / 127.0f;

  const int hi_half = (lane >> 4) & 1;   // selects feature 2*it (+1) per iteration
  const int q16 = (lane & 15) * 16;      // 16B chunk within the 256B row slice

  // Issue 22 rows x 256B = 11 full-wave b128 async gathers into `buf`.
  // Column ids are fetched with uniform scalar loads (SMEM) and lane-selected,
  // keeping the vector-memory pipe free for the async DMA gathers.
  auto prefetch = [&](int s, int base, char* buf) {
    const int* crow = cols + s * APF;    // SGPR base
#pragma unroll
    for (int it = 0; it < ASYNC_PER_SAMPLE; ++it) {
      int c0 = crow[2 * it];             // uniform -> s_load (merges into wide clauses)
      int c1 = crow[2 * it + 1];         // uniform -> s_load
      int c  = hi_half ? c1 : c0;        // one v_cndmask
      int k  = 2 * it + hi_half;         // feature row 0..21
      const float* g = weight + (size_t)c * OUT_DIM + base;
      async_copy_b128(g + (q16 >> 2), buf + k * ROW_BYTES + q16);
    }
  };

  // Force the (provably uniform) sample index into an SGPR so all per-sample
  // control data (cols/values/ply) uses the scalar memory path.
  int s = __builtin_amdgcn_readfirstlane(gw);
  int base_cur = (ply[s] / BUCKET_SIZE) * LPA;   // only the ply-bucket slice is needed
  int cur = 0;
  prefetch(s, base_cur, wbuf);

  while (true) {
    const int sn = __builtin_amdgcn_readfirstlane(s + total_waves);
    const bool has_next = (sn < m);
    int base_next = 0;
    if (has_next) {
      base_next = (ply[sn] / BUCKET_SIZE) * LPA;
      prefetch(sn, base_next, wbuf + ((cur ^ 1) * SAMPLE_BYTES));
      wait_async_keep_one_sample();
    } else {
      wait_async_all();
    }

    // Reduce 22 staged rows from LDS. Lane l owns output columns {2l, 2l+1}:
    // ds_load at byte 8*l -> banks 2l,2l+1 -> conflict-free on 64 banks.
    const char* buf = wbuf + cur * SAMPLE_BYTES;
    float ax = 0.0f, ay = 0.0f;
#pragma unroll
    for (int k = 0; k < APF; ++k) {
      const float v = values[s * APF + k];                 // SGPR index -> s_load
      const float2 w = *(const float2*)(buf + k * ROW_BYTES + lane * 8);
      ax += rintf(w.x * QS) * QI * v;                      // STE round-quant per element
      ay += rintf(w.y * QS) * QI * v;
    }
    const int c0 = base_cur + lane * 2;
    ax += rintf(bias[c0] * QS) / QS;
    ay += rintf(bias[c0 + 1] * QS) / QS;

    const float LO = (float)(-16.0 / 127.0);
    const float HI = (float)(1.0 - 16.0 / 127.0);
    const float SH = (float)(16.0 / 127.0);

    float2 r;
    {
      float x = ax;
      x = (x >= 0.0f) ? x : 0.125f * x;                    // leaky_relu(0.125)
      x = fminf(fmaxf(x, LO), HI) + SH;                    // clamp + shift
      x = fminf(fmaxf(x, 0.0f), 1.0f);                     // _fq_floor clip
      r.x = floorf(x * QS) / QS;                           // fixed-point floor quant
    }
    {
      float x = ay;
      x = (x >= 0.0f) ? x : 0.125f * x;
      x = fminf(fmaxf(x, LO), HI) + SH;
      x = fminf(fmaxf(x, 0.0f), 1.0f);
      r.y = floorf(x * QS) / QS;
    }
    *(float2*)(out + (size_t)s * LPA + lane * 2) = r;

    if (!has_next) break;
    s = sn;
    base_cur = base_next;
    cur ^= 1;
  }
}

extern "C" void kernel_launch(void* const* d_in, const int* in_sizes, int n_in,
                              void* d_out, int out_size, void* d_ws, size_t ws_size,
                              hipStream_t stream) {
  (void)n_in; (void)out_size; (void)d_ws; (void)ws_size;
  const int*   fidx   = (const int*)d_in[0];    // [2, nnz]: rows then cols
  const float* values = (const float*)d_in[1];  // [nnz]
  const int*   ply    = (const int*)d_in[4];    // [m]
  const float* weight = (const float*)d_in[5];  // [N_FEATS, OUT_DIM]
  const float* bias   = (const float*)d_in[6];  // [OUT_DIM]
  float*       out    = (float*)d_out;          // [m, LPA]

  const int nnz = in_sizes[0] / 2;
  const int m   = in_sizes[4];
  const int* cols = fidx + nnz;                 // rows are repeat(arange(m), APF) by construction

  // 2 samples per wave: double-buffer overlap + maximal wave-level parallelism
  // (in-flight async bytes per WGP comfortably cover HBM latency*BW).
  int blocks = (m + WAVES_PER_BLOCK * SAMPLES_PER_WAVE - 1) / (WAVES_PER_BLOCK * SAMPLES_PER_WAVE);
  if (blocks < 1) blocks = 1;
  const int total_waves = blocks * WAVES_PER_BLOCK;

  PhaseAdaptiveInput_kernel<<<blocks, THREADS, 0, stream>>>(
      cols, values, ply, weight, bias, out, m, total_waves);
}